// VGRNN_19645180412756
// MI455X (gfx1250) — compile-verified
//
#include <hip/hip_runtime.h>
#include <math.h>

#define HDIM 128

typedef float v2f __attribute__((ext_vector_type(2)));
typedef float v8f __attribute__((ext_vector_type(8)));

// ---------------- WMMA helpers (fp32 16x16x4, CDNA5) ----------------
__device__ __forceinline__ v8f wmma4(v2f a, v2f b, v8f c) {
  // (neg_a, A, neg_b, B, c_mod, C, reuse_a, reuse_b)
  return __builtin_amdgcn_wmma_f32_16x16x4_f32(false, a, false, b, (short)0, c, false, false);
}

// A fragment: 16x4 tile of row-major [N,128]; lane layout per ISA:
//   element(M = lane&15, K = kb + 2*(lane>>4) + vgpr)
__device__ __forceinline__ v2f load_a_frag(const float* __restrict__ M, int rowBase, int kb, int lane) {
  const int m = lane & 15;
  const int k = kb + ((lane >> 4) << 1);
  const float* p = M + (size_t)(rowBase + m) * HDIM + k;
  v2f a; a.x = p[0]; a.y = p[1];
  return a;
}

// B fragment: 4x16 tile of row-major (k-major) W[128,128]:
//   element(K = kb + 2*(lane>>4) + vgpr, N = colBase + (lane&15))
__device__ __forceinline__ v2f load_b_frag(const float* __restrict__ W, int colBase, int kb, int lane) {
  const int n = lane & 15;
  const int k = kb + ((lane >> 4) << 1);
  const float* p = W + (size_t)k * HDIM + colBase + n;
  v2f b; b.x = p[0]; b.y = p[HDIM];
  return b;
}

// C/D: VGPR j holds (M = rowBase + j + 8*(lane>>4), N = colBase + (lane&15))
__device__ __forceinline__ void store_c_frag(float* __restrict__ U, int rowBase, int colBase, int lane, v8f acc) {
  const int n = lane & 15;
  const int mo = (lane >> 4) << 3;
#pragma unroll
  for (int j = 0; j < 8; ++j)
    U[(size_t)(rowBase + mo + j) * HDIM + colBase + n] = acc[j];
}

// ---------------- fused gate GEMM: U1=X*W1x+H*W1h, U2=X*W2x+H*W2h, U3=X*W3x ----
__global__ void __launch_bounds__(128)
k_gemm3(const float* __restrict__ X, const float* __restrict__ Hm,
        const float* __restrict__ W1x, const float* __restrict__ W1h,
        const float* __restrict__ W2x, const float* __restrict__ W2h,
        const float* __restrict__ W3x,
        float* __restrict__ U1, float* __restrict__ U2, float* __restrict__ U3) {
  const int rowBase = blockIdx.x << 4;          // 16 rows per block
  const int lane = threadIdx.x & 31;
  const int wave = threadIdx.x >> 5;            // 4 waves -> 32 cols each
  const int c0 = wave << 5;
  const int c1 = c0 + 16;

  v8f a10 = {}, a11 = {}, a20 = {}, a21 = {}, a30 = {}, a31 = {};

  for (int kb = 0; kb < HDIM; kb += 4) {
    v2f ax = load_a_frag(X,  rowBase, kb, lane);
    v2f ah = load_a_frag(Hm, rowBase, kb, lane);
    v2f b;
    // interleave accumulators to space out RAW chains
    b = load_b_frag(W1x, c0, kb, lane); a10 = wmma4(ax, b, a10);
    b = load_b_frag(W2x, c0, kb, lane); a20 = wmma4(ax, b, a20);
    b = load_b_frag(W3x, c0, kb, lane); a30 = wmma4(ax, b, a30);
    b = load_b_frag(W1x, c1, kb, lane); a11 = wmma4(ax, b, a11);
    b = load_b_frag(W2x, c1, kb, lane); a21 = wmma4(ax, b, a21);
    b = load_b_frag(W3x, c1, kb, lane); a31 = wmma4(ax, b, a31);
    b = load_b_frag(W1h, c0, kb, lane); a10 = wmma4(ah, b, a10);
    b = load_b_frag(W2h, c0, kb, lane); a20 = wmma4(ah, b, a20);
    b = load_b_frag(W1h, c1, kb, lane); a11 = wmma4(ah, b, a11);
    b = load_b_frag(W2h, c1, kb, lane); a21 = wmma4(ah, b, a21);
  }
  store_c_frag(U1, rowBase, c0, lane, a10);
  store_c_frag(U1, rowBase, c1, lane, a11);
  store_c_frag(U2, rowBase, c0, lane, a20);
  store_c_frag(U2, rowBase, c1, lane, a21);
  store_c_frag(U3, rowBase, c0, lane, a30);
  store_c_frag(U3, rowBase, c1, lane, a31);
}

// ---------------- single GEMM: U4 = RH * Whh ----------------
__global__ void __launch_bounds__(128)
k_gemm1(const float* __restrict__ A, const float* __restrict__ W, float* __restrict__ U) {
  const int rowBase = blockIdx.x << 4;
  const int lane = threadIdx.x & 31;
  const int wave = threadIdx.x >> 5;
  const int c0 = wave << 5;
  const int c1 = c0 + 16;
  v8f acc0 = {}, acc1 = {};
  for (int kb = 0; kb < HDIM; kb += 4) {
    v2f a = load_a_frag(A, rowBase, kb, lane);
    v2f b0 = load_b_frag(W, c0, kb, lane);
    v2f b1 = load_b_frag(W, c1, kb, lane);
    acc0 = wmma4(a, b0, acc0);
    acc1 = wmma4(a, b1, acc1);
  }
  store_c_frag(U, rowBase, c0, lane, acc0);
  store_c_frag(U, rowBase, c1, lane, acc1);
}

// ---------------- degree / norm precompute ----------------
__global__ void k_initdeg(float* __restrict__ deg, int N) {
  int i = blockIdx.x * blockDim.x + threadIdx.x;
  if (i < N) deg[i] = 2.0f;   // self-loop weight 2 (improved=True)
}
__global__ void k_degacc(const int* __restrict__ ei, float* __restrict__ deg, int E) {
  int e = blockIdx.x * blockDim.x + threadIdx.x;
  if (e < E) atomicAdd(&deg[ei[e]], 1.0f);   // row = edge_index[0]
}
__global__ void k_dinv(float* __restrict__ deg, float* __restrict__ sn, int N) {
  int i = blockIdx.x * blockDim.x + threadIdx.x;
  if (i < N) {
    float d = deg[i];
    float di = d > 0.0f ? rsqrtf(d) : 0.0f;
    deg[i] = di;                 // deg buffer becomes dinv
    sn[i] = 2.0f * di * di;      // self-loop norm
  }
}
__global__ void k_norm(const int* __restrict__ ei, const float* __restrict__ dinv,
                       float* __restrict__ norm, int E) {
  int e = blockIdx.x * blockDim.x + threadIdx.x;
  if (e < E) norm[e] = dinv[ei[e]] * dinv[ei[E + e]];
}

// ---------------- edge scatter: P[row] += norm * U[col] (one wave/edge) ------
__global__ void __launch_bounds__(256)
k_scatter(const float* __restrict__ U, float* P,
          const int* __restrict__ ei, const float* __restrict__ norm, int E) {
  const int lane = threadIdx.x & 31;
  int wid = (blockIdx.x * blockDim.x + threadIdx.x) >> 5;
  const int nw = (gridDim.x * blockDim.x) >> 5;
  for (int e = wid; e < E; e += nw) {
    const int r = ei[e];
    const int c = ei[E + e];
    const float s = norm[e];
    float4 v = ((const float4*)(U + (size_t)c * HDIM))[lane];
    float* dst = P + (size_t)r * HDIM + (lane << 2);
    atomicAdd(dst + 0, s * v.x);
    atomicAdd(dst + 1, s * v.y);
    atomicAdd(dst + 2, s * v.z);
    atomicAdd(dst + 3, s * v.w);
  }
}

// ---------------- gate pointwise: z, r, rh (z aliases U1, rh aliases P1) -----
__global__ void k_gates(const float* P1, const float* U1, const float* P2, const float* U2,
                        const float* __restrict__ sn, const float* __restrict__ h,
                        const float* __restrict__ bz1, const float* __restrict__ bz2,
                        const float* __restrict__ br1, const float* __restrict__ br2,
                        float* zout, float* rhout, int total) {
  int i = blockIdx.x * blockDim.x + threadIdx.x;
  const int stride = gridDim.x * blockDim.x;
  for (; i < total; i += stride) {
    const int c = i & (HDIM - 1);
    const int n = i >> 7;
    const float s = sn[n];
    float z = P1[i] + s * U1[i] + bz1[c] + bz2[c];
    float r = P2[i] + s * U2[i] + br1[c] + br2[c];
    z = 1.0f / (1.0f + __expf(-z));
    r = 1.0f / (1.0f + __expf(-r));
    const float hv = h[i];
    zout[i]  = z;        // aliases U1 (same-index read-then-write: safe)
    rhout[i] = r * hv;   // aliases P1
  }
}

// ---------------- GRU update: h = z*h + (1-z)*tanh(...) ----------------------
__global__ void k_update(const float* __restrict__ P3, const float* __restrict__ U3,
                         const float* __restrict__ P4, const float* __restrict__ U4,
                         const float* __restrict__ sn, const float* __restrict__ zb,
                         float* h,
                         const float* __restrict__ bh1, const float* __restrict__ bh2,
                         float* outp, int total) {
  int i = blockIdx.x * blockDim.x + threadIdx.x;
  const int stride = gridDim.x * blockDim.x;
  for (; i < total; i += stride) {
    const int c = i & (HDIM - 1);
    const int n = i >> 7;
    const float s = sn[n];
    const float pre = P3[i] + s * U3[i] + bh1[c] + P4[i] + s * U4[i] + bh2[c];
    const float ht = tanhf(pre);
    const float z = zb[i];
    const float hv = h[i];
    const float hn = z * hv + (1.0f - z) * ht;
    h[i] = hn;
    if (outp) outp[i] = hn;
  }
}

extern "C" void kernel_launch(void* const* d_in, const int* in_sizes, int n_in,
                              void* d_out, int out_size, void* d_ws, size_t ws_size,
                              hipStream_t stream) {
  const float* x   = (const float*)d_in[0];
  const float* h0  = (const float*)d_in[1];
  const int*   ei  = (const int*)d_in[2];
  const float* Wxz = (const float*)d_in[3];
  const float* Whz = (const float*)d_in[4];
  const float* Wxr = (const float*)d_in[5];
  const float* Whr = (const float*)d_in[6];
  const float* Wxh = (const float*)d_in[7];
  const float* Whh = (const float*)d_in[8];
  const float* bxz = (const float*)d_in[9];
  const float* bhz = (const float*)d_in[10];
  const float* bxr = (const float*)d_in[11];
  const float* bhr = (const float*)d_in[12];
  const float* bxh = (const float*)d_in[13];
  const float* bhh = (const float*)d_in[14];
  float* out = (float*)d_out;

  const int L = 2, H = HDIM;
  const int E = in_sizes[2] / 2;
  const int N = in_sizes[1] / (L * H);
  const int T = (int)((long long)in_sizes[0] / ((long long)N * H));
  const size_t NH = (size_t)N * H;

  // workspace carve-out (floats)
  float* ws   = (float*)d_ws;
  float* deg  = ws;                 // N   (becomes dinv)
  float* sn   = deg + N;            // N
  float* norm = sn + N;             // E
  float* hbuf = norm + E;           // L*NH
  float* U1   = hbuf + (size_t)L * NH;
  float* U2   = U1 + NH;
  float* U3   = U2 + NH;
  float* P1   = U3 + NH;
  float* P2   = P1 + NH;
  float* P3   = P2 + NH;
  // aliases (regions dead by the time they are reused)
  float* zb = U1;   // z gate
  float* rh = P1;   // r*h
  float* U4 = U2;
  float* P4 = P2;

  // ---- per-launch precompute (edge_index is an input: recompute each call) ----
  k_initdeg<<<(N + 255) / 256, 256, 0, stream>>>(deg, N);
  k_degacc <<<(E + 255) / 256, 256, 0, stream>>>(ei, deg, E);
  k_dinv   <<<(N + 255) / 256, 256, 0, stream>>>(deg, sn, N);
  k_norm   <<<(E + 255) / 256, 256, 0, stream>>>(ei, deg, norm, E);
  hipMemcpyAsync(hbuf, h0, (size_t)L * NH * sizeof(float),
                 hipMemcpyDeviceToDevice, stream);

  const int rowTiles = N / 16;            // N = 50000 -> 3125
  const int pwGrid = 4096;
  const int scGrid = 2048;

  for (int t = 0; t < T; ++t) {
    const float* inp = x + (size_t)t * NH;
    for (int l = 0; l < L; ++l) {
      float* hl = hbuf + (size_t)l * NH;
      const size_t wo = (size_t)l * H * H;
      const size_t bo = (size_t)l * H;

      // Phase A: fused pre-messages (5 GEMMs, shared A-fragments)
      hipMemsetAsync(P1, 0, 3 * NH * sizeof(float), stream);
      k_gemm3<<<rowTiles, 128, 0, stream>>>(inp, hl,
          Wxz + wo, Whz + wo, Wxr + wo, Whr + wo, Wxh + wo, U1, U2, U3);

      // Phase B: 3 propagations (z, r combined by linearity; x-side of h~)
      k_scatter<<<scGrid, 256, 0, stream>>>(U1, P1, ei, norm, E);
      k_scatter<<<scGrid, 256, 0, stream>>>(U2, P2, ei, norm, E);
      k_scatter<<<scGrid, 256, 0, stream>>>(U3, P3, ei, norm, E);

      // gates
      k_gates<<<pwGrid, 256, 0, stream>>>(P1, U1, P2, U2, sn, hl,
          bxz + bo, bhz + bo, bxr + bo, bhr + bo, zb, rh, (int)NH);

      // Phase C: (r*h) @ Whh, propagate
      k_gemm1<<<rowTiles, 128, 0, stream>>>(rh, Whh + wo, U4);
      hipMemsetAsync(P4, 0, NH * sizeof(float), stream);
      k_scatter<<<scGrid, 256, 0, stream>>>(U4, P4, ei, norm, E);

      // GRU update (top layer also writes the output slab)
      float* outp = (l == L - 1) ? (out + (size_t)t * NH) : nullptr;
      k_update<<<pwGrid, 256, 0, stream>>>(P3, U3, P4, U4, sn, zb, hl,
          bxh + bo, bhh + bo, outp, (int)NH);

      inp = hl;   // next layer consumes the just-updated hidden state
    }
  }
}